// MessagePassing_40793599378092
// MI455X (gfx1250) — compile-verified
//
#include <hip/hip_runtime.h>
#include <hip/hip_bf16.h>

// ---------------------------------------------------------------------------
// Graph message passing: out[dst] += x[src] * w   (F = 32 == wave32 width)
// Memory-bound: edge stream 40MB (streamed once), gather/scatter both
// L2-resident (12.8MB each vs 192MB L2). Strategy: wave-per-edge
// (lane == feature), double-buffered CDNA5 async global->LDS staging of the
// edge stream (ASYNCcnt / s_wait_asynccnt), f32 global atomics for scatter.
// ---------------------------------------------------------------------------

#define BLOCK 256
#define TILE  512
#define WARPS (BLOCK / 32)
#define EPT   (TILE / BLOCK)      // edges staged per thread per tile = 2
#define OPS_PER_TILE (3 * EPT)    // async ops issued per thread per tile = 6

typedef int v2i __attribute__((ext_vector_type(2)));

#if defined(__HIP_DEVICE_COMPILE__) && __has_builtin(__builtin_amdgcn_global_load_async_to_lds_b64) && __has_builtin(__builtin_amdgcn_global_load_async_to_lds_b32)
#define HAVE_ASYNC_LDS 1
#else
#define HAVE_ASYNC_LDS 0
#endif

#if defined(__HIP_DEVICE_COMPILE__) && __has_builtin(__builtin_amdgcn_s_wait_asynccnt)
#define WAIT_ASYNC(n) __builtin_amdgcn_s_wait_asynccnt(n)
#elif defined(__HIP_DEVICE_COMPILE__)
#define WAIT_ASYNC(n) asm volatile("s_wait_asynccnt %0" :: "n"(n) : "memory")
#else
#define WAIT_ASYNC(n) ((void)0)
#endif

// Stage one tile of edge data (src ids, dst ids, weights) into LDS.
// Tail addresses are clamped in-bounds so every thread always issues exactly
// OPS_PER_TILE async ops -> uniform s_wait_asynccnt thresholds (in-order
// completion makes "wait <= OPS_PER_TILE" equal to "previous tile landed").
__device__ __forceinline__ void stage_tile(const long long* __restrict__ idx,
                                           const float* __restrict__ wgt,
                                           long base, long E,
                                           long long* sSrc, long long* sDst,
                                           float* sW)
{
    const int tid = threadIdx.x;
#pragma unroll
    for (int k = 0; k < EPT; ++k) {
        const int slot = tid + k * BLOCK;
        long e = base + slot;
        if (e >= E) e = E - 1;   // clamp: duplicate in-bounds read, ignored later
#if HAVE_ASYNC_LDS
        __builtin_amdgcn_global_load_async_to_lds_b64((v2i*)(idx + e),
                                                      (v2i*)(sSrc + slot), 0, 0);
        __builtin_amdgcn_global_load_async_to_lds_b64((v2i*)(idx + E + e),
                                                      (v2i*)(sDst + slot), 0, 0);
        __builtin_amdgcn_global_load_async_to_lds_b32((int*)(wgt + e),
                                                      (int*)(sW + slot), 0, 0);
#else
        sSrc[slot] = idx[e];
        sDst[slot] = idx[E + e];
        sW[slot]   = wgt[e];
#endif
    }
}

// Vectorized zero-init: out_size = 100000*32 floats = 800000 float4 (16B aligned).
__global__ void __launch_bounds__(BLOCK)
mp_zero_kernel(float4* __restrict__ out4, long n4)
{
    long i = (long)blockIdx.x * BLOCK + threadIdx.x;
    if (i < n4) out4[i] = make_float4(0.f, 0.f, 0.f, 0.f);
}

__global__ void __launch_bounds__(BLOCK)
mp_scatter_kernel(const float* __restrict__ x,
                  const long long* __restrict__ idx,
                  const float* __restrict__ wgt,
                  float* __restrict__ out, long E)
{
    __shared__ long long sSrc[2][TILE];
    __shared__ long long sDst[2][TILE];
    __shared__ float     sW[2][TILE];

    const long numTiles = (E + TILE - 1) / TILE;
    long tile = blockIdx.x;
    if (tile >= numTiles) return;   // whole block exits together (no barrier yet)

    // Prologue: stage first tile.
    stage_tile(idx, wgt, tile * (long)TILE, E, sSrc[0], sDst[0], sW[0]);

    const int lane = threadIdx.x & 31;  // lane == feature index (F == 32)
    const int warp = threadIdx.x >> 5;
    int cur = 0;

    for (; tile < numTiles; tile += gridDim.x) {
        const long nextTile = tile + (long)gridDim.x;
        if (nextTile < numTiles) {
            // Issue next tile's async loads into the other buffer, then wait
            // until only those remain outstanding (in-order completion =>
            // current tile's data has landed in LDS).
            stage_tile(idx, wgt, nextTile * (long)TILE, E,
                       sSrc[cur ^ 1], sDst[cur ^ 1], sW[cur ^ 1]);
            WAIT_ASYNC(OPS_PER_TILE);
        } else {
            WAIT_ASYNC(0);
        }
        __syncthreads();   // all waves' staging for current tile visible

        const long base = tile * (long)TILE;
        const long rem  = E - base;
        const int  lim  = rem < (long)TILE ? (int)rem : TILE;

        // One wave per edge: LDS broadcast of (src,dst,w); 128B coalesced
        // gather from x (L2-resident); 32 f32 global atomics covering one
        // 128B output row (L2-resident).
        for (int s = warp; s < lim; s += WARPS) {
            const long long src = sSrc[cur][s];
            const long long dst = sDst[cur][s];
            const float     w   = sW[cur][s];
            const float     v   = x[src * 32 + lane];
            atomicAdd(out + dst * 32 + lane, v * w);
        }
        __syncthreads();   // done reading buf[cur] before it is re-staged
        cur ^= 1;
    }
}

extern "C" void kernel_launch(void* const* d_in, const int* in_sizes, int n_in,
                              void* d_out, int out_size, void* d_ws, size_t ws_size,
                              hipStream_t stream)
{
    const float*     x   = (const float*)d_in[0];
    const long long* idx = (const long long*)d_in[1];   // int64 (2, E) row-major
    const float*     wgt = (const float*)d_in[2];
    float*           out = (float*)d_out;

    const long E    = (long)in_sizes[2];        // number of edges
    const long nOut = (long)out_size;           // N_NODES * 32 (multiple of 4)

    // Zero-init the accumulation target (harness poisons d_out).
    {
        const long n4 = nOut / 4;
        long blocks = (n4 + BLOCK - 1) / BLOCK;
        if (blocks < 1) blocks = 1;
        mp_zero_kernel<<<(int)blocks, BLOCK, 0, stream>>>((float4*)out, n4);
    }

    const long numTiles = (E + TILE - 1) / TILE;
    long grid = numTiles < 1024 ? numTiles : 1024;  // ~4 tiles/block -> pipeline
    if (grid < 1) grid = 1;
    mp_scatter_kernel<<<(int)grid, BLOCK, 0, stream>>>(x, idx, wgt, out, E);
}